// SAMGCN_48765058678829
// MI455X (gfx1250) — compile-verified
//
#include <hip/hip_runtime.h>
#include <hip/hip_bf16.h>

typedef __attribute__((ext_vector_type(2))) float v2f;
typedef __attribute__((ext_vector_type(8))) float v8f;

#define NNODES 50000
#define NFEAT  1500
#define NH1    512
#define NH2    128

enum { EP_NONE = 0, EP_BIAS = 1, EP_SIGMOID = 2, EP_SOFTPLUS = 3, EP_EXPCLIP = 4 };

// ---------------------------------------------------------------------------
// f32 WMMA GEMM: C[M,NCOLS] = A[M,K] @ B[K,NCOLS] (+ bias, + elementwise)
// Fully templated shapes; 32-bit element offsets off SGPR kernarg bases so
// loads lower to global_load (SADDR+VOFFSET) with constant instruction
// offsets; software-pipelined fragment double-buffering so WMMAs only wait on
// loads issued one iteration earlier. Column overflow (NCOLS=1500) handled by
// clamping load indices; invalid lanes are masked at store time.
// ---------------------------------------------------------------------------
template <int EPI, int M, int K, int NCOLS>
__global__ __launch_bounds__(128) void gemm_wmma_f32(
    const float* __restrict__ A, const float* __restrict__ B,
    const float* __restrict__ bias, float* __restrict__ C) {
  constexpr int CT = 4;
  constexpr int colSlabs = (NCOLS + CT * 16 - 1) / (CT * 16);
  constexpr int totalWaves = (M / 16) * colSlabs;

  const int wave = blockIdx.x * (blockDim.x >> 5) + (threadIdx.x >> 5);
  if (wave >= totalWaves) return;  // wave-uniform exit: EXEC all-1 for WMMA

  const int rowTile = wave / colSlabs;
  const int slab    = wave % colSlabs;
  const int rowBase = rowTile * 16;
  const int colBase = slab * (CT * 16);

  const int lane = threadIdx.x & 31;
  const int half = lane >> 4;   // selects K-pair for A/B fragments
  const int l15  = lane & 15;

  v8f acc[CT] = {};

  // 32-bit element offsets (branch-free; column index clamped into range)
  unsigned boff[CT];
#pragma unroll
  for (int t = 0; t < CT; ++t) {
    int n = colBase + t * 16 + l15;
    n = (n < NCOLS) ? n : (NCOLS - 1);
    boff[t] = (unsigned)(half * 2 * NCOLS + n);
  }
  unsigned aoff = (unsigned)((rowBase + l15) * K + half * 2);

  // Prologue: load first fragments
  v2f a;
  a.x = A[aoff];
  a.y = A[aoff + 1];
  v2f b[CT];
#pragma unroll
  for (int t = 0; t < CT; ++t) {
    b[t].x = B[boff[t]];
    b[t].y = B[boff[t] + NCOLS];
  }

#pragma unroll 2
  for (int k = 0; k < K - 4; k += 4) {
    // Issue next step's loads before consuming current fragments.
    aoff += 4;
    v2f an;
    an.x = A[aoff];
    an.y = A[aoff + 1];
    v2f bn[CT];
#pragma unroll
    for (int t = 0; t < CT; ++t) {
      boff[t] += 4 * NCOLS;
      bn[t].x = B[boff[t]];
      bn[t].y = B[boff[t] + NCOLS];
    }
#pragma unroll
    for (int t = 0; t < CT; ++t) {
      acc[t] = __builtin_amdgcn_wmma_f32_16x16x4_f32(
          false, a, false, b[t], (short)0, acc[t], false, false);
    }
    a = an;
#pragma unroll
    for (int t = 0; t < CT; ++t) b[t] = bn[t];
  }
  // Epilogue step
#pragma unroll
  for (int t = 0; t < CT; ++t) {
    acc[t] = __builtin_amdgcn_wmma_f32_16x16x4_f32(
        false, a, false, b[t], (short)0, acc[t], false, false);
  }

#pragma unroll
  for (int t = 0; t < CT; ++t) {
    const int col = colBase + t * 16 + l15;
    if (col < NCOLS) {
      float bv = (EPI != EP_NONE) ? bias[col] : 0.0f;
#pragma unroll
      for (int r = 0; r < 8; ++r) {
        const int m = half * 8 + r;
        float v = acc[t][r] + bv;
        if (EPI == EP_SIGMOID) {
          v = 1.0f / (1.0f + expf(-v));
        } else if (EPI == EP_SOFTPLUS) {
          v = (v > 20.0f) ? v : log1pf(expf(v));
          v = fminf(fmaxf(v, 1e-4f), 1e4f);
        } else if (EPI == EP_EXPCLIP) {
          v = expf(v);
          v = fminf(fmaxf(v, 1e-5f), 1e6f);
        }
        C[(unsigned)(rowBase + m) * NCOLS + col] = v;
      }
    }
  }
}

// ---------------------------------------------------------------------------
// out[i,:] = scale * b[:]  (bias broadcast, used to seed spmm accumulators)
// ---------------------------------------------------------------------------
__global__ __launch_bounds__(256) void bias_init_kernel(
    float* __restrict__ out, const float* __restrict__ b, float scale, int n) {
  int idx = blockIdx.x * blockDim.x + threadIdx.x;
  if (idx < n) out[idx] = scale * b[idx & (NH2 - 1)];
}

// ---------------------------------------------------------------------------
// spmm scatter-add: out[rows[e],:] += vals[e] * x[cols[e],:]
// One wave = one edge; lane g handles 4 features (coalesced 512B per wave).
// ---------------------------------------------------------------------------
__global__ __launch_bounds__(256) void spmm_atomic_kernel(
    const int* __restrict__ rows, const int* __restrict__ cols,
    const float* __restrict__ vals, const float* __restrict__ x,
    float* __restrict__ out, int nE) {
  long tid = (long)blockIdx.x * blockDim.x + threadIdx.x;
  int e = (int)(tid >> 5);
  if (e >= nE) return;
  int g = (int)(tid & 31);
  int r = rows[e];
  int c = cols[e];
  float v = vals[e];
  const float* xp = x + (size_t)c * NH2 + g * 4;
  float* op = out + (size_t)r * NH2 + g * 4;
#pragma unroll
  for (int i = 0; i < 4; ++i) atomicAdd(op + i, v * xp[i]);
}

// ---------------------------------------------------------------------------
// Attention: one block (128 threads) per node.
// z = [emb1, emb3+emb4, emb2]; w_j = att_w2 . tanh(z_j @ att_w1 + att_b1);
// beta = softmax(w); emb_att = sum_j beta_j z_j
// ---------------------------------------------------------------------------
__global__ __launch_bounds__(128) void attention_kernel(
    const float* __restrict__ e1, const float* __restrict__ z1b,
    const float* __restrict__ e2, const float* __restrict__ att_w1,
    const float* __restrict__ att_b1, const float* __restrict__ att_w2,
    float* __restrict__ emb_att) {
  __shared__ float z[3][NH2];
  __shared__ float h[3][16];
  __shared__ float beta[3];
  const int i = blockIdx.x;
  const int t = threadIdx.x;
  z[0][t] = e1[(size_t)i * NH2 + t];
  z[1][t] = z1b[(size_t)i * NH2 + t];
  z[2][t] = e2[(size_t)i * NH2 + t];
  __syncthreads();
  if (t < 48) {
    const int j = t >> 4, kk = t & 15;
    float s = att_b1[kk];
    for (int f = 0; f < NH2; ++f) s += z[j][f] * att_w1[f * 16 + kk];
    h[j][kk] = tanhf(s);
  }
  __syncthreads();
  if (t == 0) {
    float sc[3];
    float mx = -1e30f;
    for (int j = 0; j < 3; ++j) {
      float s = 0.0f;
      for (int kk = 0; kk < 16; ++kk) s += h[j][kk] * att_w2[kk];
      sc[j] = s;
      mx = fmaxf(mx, s);
    }
    float den = 0.0f;
    for (int j = 0; j < 3; ++j) { sc[j] = expf(sc[j] - mx); den += sc[j]; }
    for (int j = 0; j < 3; ++j) beta[j] = sc[j] / den;
  }
  __syncthreads();
  emb_att[(size_t)i * NH2 + t] =
      beta[0] * z[0][t] + beta[1] * z[1][t] + beta[2] * z[2][t];
}

// ---------------------------------------------------------------------------
// BatchNorm over rows (training mode, biased variance)
// ---------------------------------------------------------------------------
__global__ __launch_bounds__(256) void zero_kernel(float* __restrict__ p, int n) {
  int i = blockIdx.x * blockDim.x + threadIdx.x;
  if (i < n) p[i] = 0.0f;
}

__global__ __launch_bounds__(512) void bn_partial_kernel(
    const float* __restrict__ x, float* __restrict__ gsum,
    float* __restrict__ gsumsq, int nRows) {
  const int c = threadIdx.x;  // column 0..511
  const int r0 = blockIdx.x * 128;
  const int r1 = min(r0 + 128, nRows);
  float s = 0.0f, s2 = 0.0f;
  for (int r = r0; r < r1; ++r) {
    float v = x[(size_t)r * NH1 + c];
    s += v;
    s2 += v * v;
  }
  atomicAdd(&gsum[c], s);
  atomicAdd(&gsumsq[c], s2);
}

__global__ __launch_bounds__(512) void bn_finalize_kernel(
    const float* __restrict__ gsum, const float* __restrict__ gsumsq,
    const float* __restrict__ gamma, const float* __restrict__ bbeta,
    float* __restrict__ scale, float* __restrict__ shift) {
  int c = blockIdx.x * blockDim.x + threadIdx.x;
  if (c < NH1) {
    const float inv_n = 1.0f / (float)NNODES;
    float mu = gsum[c] * inv_n;
    float var = gsumsq[c] * inv_n - mu * mu;
    float sc = rsqrtf(var + 1e-5f) * gamma[c];
    scale[c] = sc;
    shift[c] = bbeta[c] - mu * sc;
  }
}

__global__ __launch_bounds__(256) void bn_apply_kernel(
    const float* __restrict__ x, const float* __restrict__ scale,
    const float* __restrict__ shift, float* __restrict__ y, long n) {
  long idx = (long)blockIdx.x * blockDim.x + threadIdx.x;
  if (idx < n) {
    int c = (int)(idx & (NH1 - 1));
    float v = x[idx] * scale[c] + shift[c];
    y[idx] = fmaxf(v, 0.0f);
  }
}

// ---------------------------------------------------------------------------
// Host-side launcher
// ---------------------------------------------------------------------------
template <int EPI, int M, int K, int NCOLS>
static void launch_gemm(const float* A, const float* B, const float* bias,
                        float* C, hipStream_t stream) {
  constexpr int CT = 4;
  constexpr int colSlabs = (NCOLS + CT * 16 - 1) / (CT * 16);
  constexpr int totalWaves = (M / 16) * colSlabs;
  constexpr int blocks = (totalWaves + 3) / 4;
  gemm_wmma_f32<EPI, M, K, NCOLS><<<blocks, 128, 0, stream>>>(A, B, bias, C);
}

extern "C" void kernel_launch(void* const* d_in, const int* in_sizes, int n_in,
                              void* d_out, int out_size, void* d_ws, size_t ws_size,
                              hipStream_t stream) {
  const float* oridata = (const float*)d_in[0];
  const float* augdata = (const float*)d_in[1];
  const int*   stg_rows = (const int*)d_in[2];
  const int*   stg_cols = (const int*)d_in[3];
  const float* stg_vals = (const float*)d_in[4];
  const int*   nf_rows  = (const int*)d_in[5];
  const int*   nf_cols  = (const int*)d_in[6];
  const float* nf_vals  = (const float*)d_in[7];
  const int*   ns_rows  = (const int*)d_in[8];
  const int*   ns_cols  = (const int*)d_in[9];
  const float* ns_vals  = (const float*)d_in[10];
  const float* W2       = (const float*)d_in[11];
  const float* b2       = (const float*)d_in[12];
  const float* att_w1   = (const float*)d_in[13];
  const float* att_b1   = (const float*)d_in[14];
  const float* att_w2   = (const float*)d_in[15];
  const float* mlp_w    = (const float*)d_in[16];
  const float* mlp_b    = (const float*)d_in[17];
  const float* dec_w    = (const float*)d_in[18];
  const float* dec_b    = (const float*)d_in[19];
  const float* bn_gamma = (const float*)d_in[20];
  const float* bn_beta  = (const float*)d_in[21];
  const float* pi_w     = (const float*)d_in[22];
  const float* pi_b     = (const float*)d_in[23];
  const float* disp_w   = (const float*)d_in[24];
  const float* disp_b   = (const float*)d_in[25];
  const float* mean_w   = (const float*)d_in[26];
  const float* mean_b   = (const float*)d_in[27];

  const int nE = in_sizes[2];  // 800000

  // ---- workspace layout (floats) ----
  float* ws = (float*)d_ws;
  size_t off = 0;
  float* sup_ori = ws + off; off += (size_t)NNODES * NH2;
  float* sup_aug = ws + off; off += (size_t)NNODES * NH2;
  float* z1buf   = ws + off; off += (size_t)NNODES * NH2;
  float* emb_att = ws + off; off += (size_t)NNODES * NH2;
  float* x_pre   = ws + off; off += (size_t)NNODES * NH1;
  float* x_bn    = ws + off; off += (size_t)NNODES * NH1;
  float* gsum    = ws + off; off += NH1;
  float* gsumsq  = ws + off; off += NH1;
  float* bscale  = ws + off; off += NH1;
  float* bshift  = ws + off; off += NH1;

  // ---- output layout (concatenated, return order) ----
  float* out = (float*)d_out;
  float* emb1O = out;
  float* emb2O = out + (size_t)NNODES * NH2;
  float* embO  = out + (size_t)NNODES * NH2 * 2;
  float* piO   = out + (size_t)NNODES * NH2 * 3;
  float* dispO = piO + (size_t)NNODES * NFEAT;
  float* meanO = dispO + (size_t)NNODES * NFEAT;

  // 1) support = X @ W2 for ori and aug  (WMMA f32)
  launch_gemm<EP_NONE, NNODES, NH1, NH2>(oridata, W2, nullptr, sup_ori, stream);
  launch_gemm<EP_NONE, NNODES, NH1, NH2>(augdata, W2, nullptr, sup_aug, stream);

  // 2) seed spmm accumulators with bias (z1 gets 2*b2 since emb3+emb4)
  {
    int n = NNODES * NH2;
    int blocks = (n + 255) / 256;
    bias_init_kernel<<<blocks, 256, 0, stream>>>(emb1O, b2, 1.0f, n);
    bias_init_kernel<<<blocks, 256, 0, stream>>>(emb2O, b2, 1.0f, n);
    bias_init_kernel<<<blocks, 256, 0, stream>>>(z1buf, b2, 2.0f, n);
  }

  // 3) spmm scatter-adds
  {
    long tthreads = (long)nE * 32;
    int blocks = (int)((tthreads + 255) / 256);
    spmm_atomic_kernel<<<blocks, 256, 0, stream>>>(stg_rows, stg_cols, stg_vals, sup_ori, emb1O, nE);
    spmm_atomic_kernel<<<blocks, 256, 0, stream>>>(stg_rows, stg_cols, stg_vals, sup_aug, emb2O, nE);
    spmm_atomic_kernel<<<blocks, 256, 0, stream>>>(nf_rows, nf_cols, nf_vals, sup_ori, z1buf, nE);
    spmm_atomic_kernel<<<blocks, 256, 0, stream>>>(ns_rows, ns_cols, ns_vals, sup_ori, z1buf, nE);
  }

  // 4) attention -> emb_att
  attention_kernel<<<NNODES, 128, 0, stream>>>(emb1O, z1buf, emb2O, att_w1,
                                               att_b1, att_w2, emb_att);

  // 5) emb = emb_att @ mlp_w + mlp_b   (output slot 2)
  launch_gemm<EP_BIAS, NNODES, NH2, NH2>(emb_att, mlp_w, mlp_b, embO, stream);

  // 6) x_pre = emb @ dec_w + dec_b
  launch_gemm<EP_BIAS, NNODES, NH2, NH1>(embO, dec_w, dec_b, x_pre, stream);

  // 7) batch-norm (training stats) + ReLU -> x_bn
  zero_kernel<<<(2 * NH1 + 255) / 256, 256, 0, stream>>>(gsum, 2 * NH1);  // gsum+gsumsq contiguous
  bn_partial_kernel<<<(NNODES + 127) / 128, 512, 0, stream>>>(x_pre, gsum, gsumsq, NNODES);
  bn_finalize_kernel<<<1, 512, 0, stream>>>(gsum, gsumsq, bn_gamma, bn_beta, bscale, bshift);
  {
    long n = (long)NNODES * NH1;
    int blocks = (int)((n + 255) / 256);
    bn_apply_kernel<<<blocks, 256, 0, stream>>>(x_pre, bscale, bshift, x_bn, n);
  }

  // 8) ZINB heads (big WMMA GEMMs with fused epilogues)
  launch_gemm<EP_SIGMOID,  NNODES, NH1, NFEAT>(x_bn, pi_w,   pi_b,   piO,   stream);
  launch_gemm<EP_SOFTPLUS, NNODES, NH1, NFEAT>(x_bn, disp_w, disp_b, dispO, stream);
  launch_gemm<EP_EXPCLIP,  NNODES, NH1, NFEAT>(x_bn, mean_w, mean_b, meanO, stream);
}